// MHFSpectralConvV2_29721173688950
// MI455X (gfx1250) — compile-verified
//
#include <hip/hip_runtime.h>
#include <math.h>

// ---------------------------------------------------------------------------
// MHFSpectralConv on gfx1250 (MI455X): mode-truncated DFT as f32 WMMA GEMMs,
// with Tensor-Data-Mover (TDM) staging of all dense tiles into LDS.
//   phase1: per input image (1024 WGs): P = x @ Fc  (LDS),  M = T{1,2} @ Pst
//   phase2: per (h,kx,ky) site (2176 waves): complex 16x16x16 channel mix
//   phase3: per output image (1024 WGs): U = Ex @ Ystack, out = U @ G
// Matrix math: V_WMMA_F32_16X16X4_F32.  Staging: TENSOR_LOAD_TO_LDS with
// D# pad fields generating the padded (bank-conflict-free) LDS strides.
// ---------------------------------------------------------------------------

typedef float v2f __attribute__((ext_vector_type(2)));
typedef float v8f __attribute__((ext_vector_type(8)));
typedef unsigned int u32x4 __attribute__((ext_vector_type(4)));
typedef int i32x4 __attribute__((ext_vector_type(4)));
typedef int i32x8 __attribute__((ext_vector_type(8)));

// workspace layout (float offsets)
#define WS_FC    0u                     // [256][64]  forward W-transform factors
#define WS_T1    16384u                 // [32][512]  [cos | sin]
#define WS_T2    32768u                 // [32][512]  [-sin | cos]
#define WS_EX    49152u                 // [256][64]  inverse H-transform [cos|sin]/256
#define WS_GT    65536u                 // [64][256]  inverse W-transform (irfft weights)
#define WS_XRE   81920u                 // [1024][32][17] forward modes (re)
#define WS_XIM   (81920u + 557056u)     // forward modes (im)
#define WS_YRE   (81920u + 2u*557056u)  // mixed modes (re)
#define WS_YIM   (81920u + 3u*557056u)  // mixed modes (im)

__device__ __forceinline__ v8f wmma4(v2f a, v2f b, v8f c) {
  // D = A(16x4) * B(4x16) + C, full f32
  return __builtin_amdgcn_wmma_f32_16x16x4_f32(false, a, false, b, (short)0, c,
                                               false, false);
}

// A-tile loader: row-major matrix, A is 16x4 (M x K).
// lanes 0-15: M=lane, VGPR0=K0, VGPR1=K1 ; lanes 16-31: K2/K3.
template <typename P>
__device__ __forceinline__ v2f load_a(P p, int stride, int m0, int k0) {
  int lane = threadIdx.x & 31;
  int m = lane & 15, hf = lane >> 4;
  v2f r;
  r.x = p[(m0 + m) * stride + k0 + 2 * hf];
  r.y = p[(m0 + m) * stride + k0 + 2 * hf + 1];
  return r;
}

// B-tile loader: row-major matrix, B is 4x16 (K x N), row-striped per VGPR.
template <typename P>
__device__ __forceinline__ v2f load_b(P p, int stride, int k0, int n0) {
  int lane = threadIdx.x & 31;
  int n = lane & 15, hf = lane >> 4;
  v2f r;
  r.x = p[(k0 + 2 * hf) * stride + n0 + n];
  r.y = p[(k0 + 2 * hf + 1) * stride + n0 + n];
  return r;
}

// ---------------------------------------------------------------------------
// TDM helpers: generic->LDS address, and a 2D tile-load descriptor builder.
// D# layout per CDNA5 ISA.
// pad_interval code: 0=2,1=4,2=8,3=16,4=32,5=64,6=128,7=256 DWORDs
// pad_amount  code: n -> (n+1) DWORDs of padding
// ---------------------------------------------------------------------------
__device__ __forceinline__ unsigned lds_off(const void* p) {
  return (unsigned)(unsigned long long)(__attribute__((address_space(3))) const char*)p;
}

__device__ __forceinline__ void tdm_load_2d(unsigned lds_byte_off, const void* gaddr,
                                            unsigned tile_d0, unsigned tile_d1,
                                            unsigned tens_d0, unsigned tens_d1,
                                            unsigned stride0_elems,
                                            unsigned pad_interval_code,
                                            unsigned pad_amount_code) {
  const unsigned long long ga = (unsigned long long)gaddr;
  u32x4 g0;
  g0[0] = 1u;                                          // count=1 (valid), user mode
  g0[1] = lds_byte_off;                                // lds_addr [63:32]
  g0[2] = (unsigned)(ga & 0xFFFFFFFFu);                // global_addr low
  g0[3] = (unsigned)((ga >> 32) & 0x01FFFFFFu) | (2u << 30);  // addr hi | type=2
  i32x8 g1;
  g1[0] = (int)((2u << 16)                             // data_size = 4B
                | (1u << 20)                           // pad_enable
                | (pad_interval_code << 22)
                | (pad_amount_code << 25));
  g1[1] = (int)(tens_d0 << 16);                        // tensor_dim0[15:0] @ [63:48]
  g1[2] = (int)((tens_d0 >> 16) | (tens_d1 << 16));    // dim0 hi | tensor_dim1 lo
  g1[3] = (int)((tens_d1 >> 16) | (tile_d0 << 16));    // dim1 hi | tile_dim0
  g1[4] = (int)(tile_d1 & 0xFFFFu);                    // tile_dim1 ; tile_dim2=0 (2D)
  g1[5] = (int)stride0_elems;                          // tensor_dim0_stride lo32
  g1[6] = 0;                                           // stride0 hi | stride1 lo
  g1[7] = 0;                                           // stride1 hi (unused, 2D)
  const i32x4 z4 = {0, 0, 0, 0};                       // groups 2/3 unused (2D)
  const i32x8 z8 = {0, 0, 0, 0, 0, 0, 0, 0};
  // clang-23 / therock-10.0 toolchain: 6-arg form
  __builtin_amdgcn_tensor_load_to_lds(g0, g1, z4, z4, z8, 0);
}

// ---------------------------------------------------------------------------
// Table init: exact phase reduction (k*n) & 255 keeps sinf/cosf args in [0,2pi)
// ---------------------------------------------------------------------------
__global__ __launch_bounds__(256) void init_tables(float* __restrict__ ws) {
  const int idx = blockIdx.x * 256 + threadIdx.x;   // 0..16383
  const float STEP = 6.283185307179586476925f / 256.0f;
  if (idx >= 16384) return;
  {  // Fc [w][c]: c<32 -> cos(2pi ky w/256) ; c>=32 -> -sin ; zero for ky>=17
    int w = idx >> 6, c = idx & 63, ky = c & 31;
    float v = 0.f;
    if (ky < 17) {
      float ang = STEP * (float)((ky * w) & 255);
      v = (c < 32) ? cosf(ang) : -sinf(ang);
    }
    ws[WS_FC + idx] = v;
  }
  {  // T1 = [cos | sin], T2 = [-sin | cos]  (32 x 512)
    int kx = idx >> 9, j = idx & 511, hh = j & 255;
    float ang = STEP * (float)((kx * hh) & 255);
    float cs = cosf(ang), sn = sinf(ang);
    ws[WS_T1 + idx] = (j < 256) ? cs : sn;
    ws[WS_T2 + idx] = (j < 256) ? -sn : cs;
  }
  {  // Excat [r][c]: c<32 -> cos(2pi kx r/256)/256 ; c>=32 -> sin(...)/256
    int r = idx >> 6, c = idx & 63, kx = c & 31;
    float ang = STEP * (float)((kx * r) & 255);
    ws[WS_EX + idx] = ((c < 32) ? cosf(ang) : sinf(ang)) * (1.0f / 256.0f);
  }
  {  // Gt [row][w]: row<32 -> g*cos(2pi ky w/256) ; row>=32 -> -g*sin ; g=(1|2)/256
    int row = idx >> 8, w = idx & 255, ky = row & 31;
    float v = 0.f;
    if (ky < 17) {
      float g = (ky == 0 ? 1.0f : 2.0f) * (1.0f / 256.0f);
      float ang = STEP * (float)((ky * w) & 255);
      v = (row < 32) ? g * cosf(ang) : -g * sinf(ang);
    }
    ws[WS_GT + idx] = v;
  }
}

// ---------------------------------------------------------------------------
// Phase 1: one WG per input image (b,h,i).  G1: P = x @ Fc,  G2: M = T @ Pst
// x-chunk and Fc-chunk staged by TDM with pad -> 68-dword LDS stride.
// ---------------------------------------------------------------------------
__global__ __launch_bounds__(256) void phase1_fwd(const float* __restrict__ x,
                                                  float* __restrict__ ws) {
  __shared__ float lx[256][68];    // x column-chunk (padded stride: bank spread)
  __shared__ float lfc[64][68];    // Fc row-chunk
  __shared__ float pst[512][36];   // stacked [Pre ; Pim], 32 ky cols (+pad)

  const float* Fc = ws + WS_FC;
  const float* T1 = ws + WS_T1;
  const float* T2 = ws + WS_T2;
  float* Xre = ws + WS_XRE;
  float* Xim = ws + WS_XIM;

  const int ic = blockIdx.x;                       // b*64 + h*16 + i
  const float* img = x + (size_t)ic * 65536u;
  const int tid = threadIdx.x;
  const int wv = tid >> 5;                         // 8 waves
  const int m0a = (wv * 2) * 16, m0b = m0a + 16;   // two M-tiles per wave

  v8f acc[2][4];
#pragma unroll
  for (int t = 0; t < 2; ++t)
#pragma unroll
    for (int j = 0; j < 4; ++j) acc[t][j] = (v8f){};

  for (int ch = 0; ch < 4; ++ch) {
    const int k0g = ch * 64;
    __syncthreads();  // previous chunk fully consumed by all waves
    if (wv == 0) {
      // x chunk: 64x256 dwords, row stride 256, pad 64dw interval / +4dw
      tdm_load_2d(lds_off(&lx[0][0]), img + k0g,
                  /*tile*/ 64, 256, /*tensor*/ 256, 256, /*stride0*/ 256,
                  /*pad_interval 64dw*/ 5, /*pad +4dw*/ 3);
      // Fc chunk: 64x64 dwords (contiguous rows), same pad -> stride 68
      tdm_load_2d(lds_off(&lfc[0][0]), Fc + k0g * 64,
                  64, 64, 64, 64, 64, 5, 3);
      if (ch < 3)  // warm L2 for next chunk while TDM drains this one
        __builtin_prefetch(img + (tid & 15) * 1024 + k0g + 64, 0, 1);
      __builtin_amdgcn_s_wait_tensorcnt(0);
    }
    __syncthreads();
    // G1 partial: M=256(h) N=64([Pre|Pim]) K=64(w-chunk)
    for (int ks = 0; ks < 64; ks += 4) {
      v2f a0 = load_a(&lx[0][0], 68, m0a, ks);
      v2f a1 = load_a(&lx[0][0], 68, m0b, ks);
#pragma unroll
      for (int j = 0; j < 4; ++j) {
        v2f b = load_b(&lfc[0][0], 68, ks, j * 16);
        acc[0][j] = wmma4(a0, b, acc[0][j]);
        acc[1][j] = wmma4(a1, b, acc[1][j]);
      }
    }
  }
  __syncthreads();
  // scatter P tiles into stacked form: rows 0..255 = Pre, 256..511 = Pim
  {
    const int lane = tid & 31;
    const int n = lane & 15, hf = lane >> 4;
#pragma unroll
    for (int t = 0; t < 2; ++t) {
      const int m0 = t ? m0b : m0a;
#pragma unroll
      for (int j = 0; j < 4; ++j) {
        const int c0 = j * 16 + n;
#pragma unroll
        for (int e = 0; e < 8; ++e) {
          const int row = m0 + e + 8 * hf;
          const float v = acc[t][j][e];
          if (c0 < 32) pst[row][c0] = v;
          else         pst[256 + row][c0 - 32] = v;
        }
      }
    }
  }
  __syncthreads();
  // G2: M_re = T1 @ Pst, M_im = T2 @ Pst.  8 C-tiles, one per wave. K = 512.
  {
    const int g = wv >> 2, mt = (wv >> 1) & 1, nt = wv & 1;
    const float* T = g ? T2 : T1;
    v8f c = (v8f){};
    for (int ks = 0; ks < 512; ks += 4) {
      v2f a = load_a(T, 512, mt * 16, ks);
      v2f b = load_b(&pst[0][0], 36, ks, nt * 16);
      c = wmma4(a, b, c);
    }
    float* dst = g ? Xim : Xre;
    const int lane = tid & 31;
    const int n = lane & 15, hf = lane >> 4;
    const int ky = nt * 16 + n;
    if (ky < 17) {
#pragma unroll
      for (int e = 0; e < 8; ++e) {
        const int kx = mt * 16 + e + 8 * hf;
        dst[((size_t)ic * 32 + kx) * 17 + ky] = c[e];
      }
    }
  }
}

// ---------------------------------------------------------------------------
// Phase 2: complex 16x16x16 channel mix per (h,kx,ky).  One site per wave.
// [Ore|Oim](16x32) = [Xre|Xim](16x32) @ [[Wre,Wim],[-Wim,Wre]](32x32)
// ---------------------------------------------------------------------------
__global__ __launch_bounds__(256) void phase2_mix(const float* __restrict__ wre,
                                                  const float* __restrict__ wim,
                                                  float* __restrict__ ws) {
  const float* Xre = ws + WS_XRE;
  const float* Xim = ws + WS_XIM;
  float* Yre = ws + WS_YRE;
  float* Yim = ws + WS_YIM;

  const int wv = threadIdx.x >> 5;
  const int s = blockIdx.x * 8 + wv;      // 272*8 == 2176 sites exactly
  const int h = s / 544;
  const int rem = s - h * 544;
  const int kx = rem / 17;
  const int ky = rem - kx * 17;
  const int lane = threadIdx.x & 31;
  const int lm = lane & 15, hf = lane >> 4;

  v8f acc0 = (v8f){}, acc1 = (v8f){};
#pragma unroll
  for (int ks = 0; ks < 32; ks += 4) {
    const int k = ks + 2 * hf;            // even; k,k+1 never straddle 16
    // A: Xcat[b=lm][k]  (k<16 -> Xre[i=k], else Xim[i=k-16])
    v2f a;
    {
      const float* p = (k < 16) ? Xre : Xim;
      const size_t base = (size_t)((lm * 4 + h) * 16 + (k & 15)) * 32;
      a.x = p[(base + kx) * 17 + ky];
      a.y = p[(base + 32 + kx) * 17 + ky];   // i+1 -> +32 in kx-major flat index
    }
    // B: Wstack rows k,k+1 ; cols: nt0 -> (Wre / -Wim), nt1 -> (Wim / Wre)
    v2f b0, b1;
    {
      const bool top = (k < 16);
      const int i0 = k & 15, i1 = (k + 1) & 15, o = lm;
      const size_t wb = (size_t)(h * 16);
      const size_t ia = (((wb + i0) * 16 + o) * 32 + kx) * 17 + ky;
      const size_t ib = (((wb + i1) * 16 + o) * 32 + kx) * 17 + ky;
      b0.x = top ? wre[ia] : -wim[ia];
      b0.y = top ? wre[ib] : -wim[ib];
      b1.x = top ? wim[ia] : wre[ia];
      b1.y = top ? wim[ib] : wre[ib];
    }
    acc0 = wmma4(a, b0, acc0);
    acc1 = wmma4(a, b1, acc1);
  }
#pragma unroll
  for (int e = 0; e < 8; ++e) {
    const int b = e + 8 * hf, o = lm;
    const size_t idx = (((size_t)((b * 4 + h) * 16 + o)) * 32 + kx) * 17 + ky;
    Yre[idx] = acc0[e];
    Yim[idx] = acc1[e];
  }
}

// ---------------------------------------------------------------------------
// Phase 3: one WG per output image (b,h,o).  G3: U = Ex @ Ystack (LDS),
// G4: out = [Ure|Uim] @ G  -> global, + bias.  G table staged by TDM,
// overlapped with the (computed) Ystack fill.
// ---------------------------------------------------------------------------
__global__ __launch_bounds__(256) void phase3_inv(const float* __restrict__ bias,
                                                  float* __restrict__ out,
                                                  const float* __restrict__ ws) {
  __shared__ float ust[256][68];   // [Ure | Uim] (cols 0..63)
  __shared__ float yst[64][68];    // [[Yre,Yim],[-Yim,Yre]] stacked (64x64)
  __shared__ float gtl[64][260];   // G table staged once per WG via TDM

  const float* Ex = ws + WS_EX;
  const float* Gt = ws + WS_GT;
  const float* Yre = ws + WS_YRE;
  const float* Yim = ws + WS_YIM;

  const int oc = blockIdx.x;       // b*64 + h*16 + o
  const int tid = threadIdx.x;
  const int wv = tid >> 5;

  if (wv == 0) {
    // G: 256x64 dwords contiguous, pad every 256dw by +4dw -> stride 260
    tdm_load_2d(lds_off(&gtl[0][0]), Gt,
                /*tile*/ 256, 64, /*tensor*/ 256, 64, /*stride0*/ 256,
                /*pad_interval 256dw*/ 7, /*pad +4dw*/ 3);
  }
  {  // build Ystack (ky>=17 cols are zero) -- overlaps the TDM transfer
#pragma unroll
    for (int rr = 0; rr < 16; ++rr) {
      const int e = tid * 16 + rr;          // 4096 = 64*64
      const int row = e >> 6, col = e & 63;
      const int kx = row & 31, ky = col & 31;
      float v = 0.f;
      if (ky < 17) {
        const size_t idx = (((size_t)oc) * 32 + kx) * 17 + ky;
        const float re = Yre[idx], im = Yim[idx];
        v = (row < 32) ? ((col < 32) ? re : im) : ((col < 32) ? -im : re);
      }
      yst[row][col] = v;
    }
  }
  if (wv == 0) __builtin_amdgcn_s_wait_tensorcnt(0);
  __syncthreads();
  // G3: M=256(r) N=64 K=64 ; A=Excat(global,L2), B=yst(LDS) ; 8 tiles/wave
  {
    const int m0a = (wv * 2) * 16;
#pragma unroll
    for (int t = 0; t < 2; ++t) {
      const int m0 = m0a + t * 16;
#pragma unroll
      for (int nt = 0; nt < 4; ++nt) {
        v8f c = (v8f){};
#pragma unroll
        for (int ks = 0; ks < 64; ks += 4) {
          v2f a = load_a(Ex, 64, m0, ks);
          v2f b = load_b(&yst[0][0], 68, ks, nt * 16);
          c = wmma4(a, b, c);
        }
        const int lane = tid & 31;
        const int n = lane & 15, hf = lane >> 4;
#pragma unroll
        for (int e = 0; e < 8; ++e)
          ust[m0 + e + 8 * hf][nt * 16 + n] = c[e];
      }
    }
  }
  __syncthreads();
  // G4: M=256(r) N=256(w) K=64 ; A=ust(LDS, hoisted), B=gtl(LDS) ; 32 tiles/wave
  {
    const int cch = oc & 63;
    const float bv = bias[cch];
    float* obase = out + (size_t)oc * 65536u;
    const int m0a = (wv * 2) * 16;
#pragma unroll
    for (int t = 0; t < 2; ++t) {
      const int m0 = m0a + t * 16;
      v2f av[16];
#pragma unroll
      for (int ks = 0; ks < 16; ++ks) av[ks] = load_a(&ust[0][0], 68, m0, ks * 4);
      for (int nt = 0; nt < 16; ++nt) {
        v8f c = (v8f){};
#pragma unroll
        for (int ks = 0; ks < 16; ++ks) {
          v2f b = load_b(&gtl[0][0], 260, ks * 4, nt * 16);
          c = wmma4(av[ks], b, c);
        }
        const int lane = tid & 31;
        const int n = lane & 15, hf = lane >> 4;
        const int w = nt * 16 + n;
#pragma unroll
        for (int e = 0; e < 8; ++e) {
          const int r = m0 + e + 8 * hf;
          obase[r * 256 + w] = c[e] + bv;
        }
      }
    }
  }
}

// ---------------------------------------------------------------------------
extern "C" void kernel_launch(void* const* d_in, const int* in_sizes, int n_in,
                              void* d_out, int out_size, void* d_ws, size_t ws_size,
                              hipStream_t stream) {
  const float* x    = (const float*)d_in[0];
  const float* wre  = (const float*)d_in[1];
  const float* wim  = (const float*)d_in[2];
  const float* bias = (const float*)d_in[3];
  float* out = (float*)d_out;
  float* ws  = (float*)d_ws;   // needs ~9.3 MB

  init_tables<<<64,  256, 0, stream>>>(ws);
  phase1_fwd <<<1024, 256, 0, stream>>>(x, ws);
  phase2_mix <<<272, 256, 0, stream>>>(wre, wim, ws);
  phase3_inv <<<1024, 256, 0, stream>>>(bias, out, ws);
}